// AverageCombiner_62886911148522
// MI455X (gfx1250) — compile-verified
//
#include <hip/hip_runtime.h>
#include <hip/hip_bf16.h>
#include <stdint.h>

#define COMBINE_FRONT 1
#define COMBINE_END   2

typedef float v4f __attribute__((ext_vector_type(4)));

// ---------------------------------------------------------------------------
// Pass 1: single-block scan over flattened labels.
//  - seg_start[k] = token index of FRONT #k (k = fronts_seen_before)
//  - seg_end[k]   = token index of END   #k
//  Unassigned segments: start=0, end=-1  -> count 0 -> NaN output (matches 0/0)
// ---------------------------------------------------------------------------
__global__ void __launch_bounds__(1024)
scan_kernel(const int* __restrict__ labels, int nTok,
            int* __restrict__ seg_start, int* __restrict__ seg_end,
            int num_segments)
{
    const int T   = 1024;
    const int tid = threadIdx.x;

    // init segment tables (poisoned workspace -> must init every call)
    for (int i = tid; i < num_segments; i += T) {
        seg_start[i] = 0;
        seg_end[i]   = -1;
    }

    __shared__ int sf[1024];
    __shared__ int se[1024];

    const int per = (nTok + T - 1) / T;
    const int b   = tid * per;
    const int e   = (b + per < nTok) ? (b + per) : nTok;

    int cf = 0, ce = 0;
    for (int t = b; t < e; ++t) {
        int l = labels[t];
        cf += (l == COMBINE_FRONT);
        ce += (l == COMBINE_END);
    }
    sf[tid] = cf;
    se[tid] = ce;
    __syncthreads();

    // Hillis-Steele inclusive scan over per-thread counts
    for (int off = 1; off < T; off <<= 1) {
        int vf = (tid >= off) ? sf[tid - off] : 0;
        int ve = (tid >= off) ? se[tid - off] : 0;
        __syncthreads();
        sf[tid] += vf;
        se[tid] += ve;
        __syncthreads();
    }

    int fbase = sf[tid] - cf;   // exclusive prefix of fronts
    int ebase = se[tid] - ce;   // exclusive prefix of ends

    for (int t = b; t < e; ++t) {
        int l = labels[t];
        if (l == COMBINE_FRONT) {
            if (fbase < num_segments) seg_start[fbase] = t;
            ++fbase;
        }
        if (l == COMBINE_END) {
            if (ebase < num_segments) seg_end[ebase] = t;
            ++ebase;
        }
    }
}

// ---------------------------------------------------------------------------
// Pass 2: one block per segment. Stage member rows into LDS with the CDNA5
// async global->LDS path (global_load_async_to_lds_b128, ASYNCcnt, TH=NT to
// avoid rinsing L2 with 256MB of single-use stream data), then average.
// Each thread reads back only LDS slots it staged itself, so only
// s_wait_asynccnt is needed (no workgroup barrier). Output stored with
// non-temporal hint (written once, never re-read).
// Assumes dim % 4 == 0 and dim <= 8192 (dim4 <= 2048).
// ---------------------------------------------------------------------------
__global__ void __launch_bounds__(256)
seg_avg_async(const float4* __restrict__ enc4,
              const int* __restrict__ seg_start,
              const int* __restrict__ seg_end,
              float4* __restrict__ out4,
              int dim4)
{
    __shared__ float4 smem[2048];          // 32 KB staging buffer

    const int seg = blockIdx.x;
    const int tid = threadIdx.x;

    const int start = seg_start[seg];
    const int end   = seg_end[seg];
    int cnt = end - start + 1;
    if (cnt < 0) cnt = 0;
    const float inv = 1.0f / (float)cnt;   // inf when cnt==0 -> 0*inf = NaN

    int rows_chunk = 2048 / dim4;          // rows that fit in the 32KB buffer
    if (rows_chunk < 1) rows_chunk = 1;
    if (rows_chunk > 8) rows_chunk = 8;    // bound ASYNCcnt pressure

    float4 acc[8];                         // supports dim4 up to 8*256 = 2048
    #pragma unroll
    for (int s = 0; s < 8; ++s) acc[s] = make_float4(0.f, 0.f, 0.f, 0.f);

    for (int base = 0; base < cnt; base += rows_chunk) {
        int rows = cnt - base;
        if (rows > rows_chunk) rows = rows_chunk;

        // issue async copies: global -> LDS, 16 bytes per lane per row,
        // non-temporal (single-use stream; don't pollute L2)
        for (int r = 0; r < rows; ++r) {
            const float4* gsrc = enc4 + (size_t)(start + base + r) * (size_t)dim4;
            for (int v = tid; v < dim4; v += 256) {
                unsigned lds_addr = (unsigned)(uintptr_t)(&smem[r * dim4 + v]);
                const float4* gp = gsrc + v;
                asm volatile("global_load_async_to_lds_b128 %0, %1, off th:TH_LOAD_NT"
                             :
                             : "v"(lds_addr), "v"(gp)
                             : "memory");
            }
        }

        // wait for this wave's async transfers to land in LDS
        asm volatile("s_wait_asynccnt 0x0" ::: "memory");

        // accumulate from LDS (each thread reads only its own slots; the
        // VALU consumers force the s_wait_dscnt before any later reuse)
        for (int r = 0; r < rows; ++r) {
            int s = 0;
            for (int v = tid; v < dim4; v += 256, ++s) {
                float4 d = smem[r * dim4 + v];
                acc[s].x += d.x;
                acc[s].y += d.y;
                acc[s].z += d.z;
                acc[s].w += d.w;
            }
        }
    }

    v4f* o = (v4f*)(out4 + (size_t)seg * (size_t)dim4);
    int s = 0;
    for (int v = tid; v < dim4; v += 256, ++s) {
        v4f r;
        r.x = acc[s].x * inv;
        r.y = acc[s].y * inv;
        r.z = acc[s].z * inv;
        r.w = acc[s].w * inv;
        __builtin_nontemporal_store(r, o + v);
    }
}

// ---------------------------------------------------------------------------
// Host launcher
// Inputs (setup_inputs order):
//   0: encoded        [bs, L, dim]  float32
//   1: lengths        [bs]          int     (unused by reference)
//   2: combine_labels [bs, L]       int
//   3: lang_id        scalar        int     (unused)
//   4: num_segments   scalar        int     (recomputed from out_size)
// Output: [num_segments, dim] float32
// ---------------------------------------------------------------------------
extern "C" void kernel_launch(void* const* d_in, const int* in_sizes, int n_in,
                              void* d_out, int out_size, void* d_ws, size_t ws_size,
                              hipStream_t stream)
{
    const float* enc    = (const float*)d_in[0];
    const int*   labels = (const int*)d_in[2];

    const int nTok = in_sizes[2];               // bs * L
    const int dim  = in_sizes[0] / nTok;        // 1024
    const int dim4 = dim / 4;
    const int num_segments = out_size / dim;    // 8192

    int* seg_start = (int*)d_ws;
    int* seg_end   = seg_start + num_segments;

    hipLaunchKernelGGL(scan_kernel, dim3(1), dim3(1024), 0, stream,
                       labels, nTok, seg_start, seg_end, num_segments);

    hipLaunchKernelGGL(seg_avg_async, dim3(num_segments), dim3(256), 0, stream,
                       (const float4*)enc, seg_start, seg_end,
                       (float4*)d_out, dim4);
}